// MultiHeadAttention_60275571032125
// MI455X (gfx1250) — compile-verified
//
#include <hip/hip_runtime.h>
#include <hip/hip_bf16.h>

typedef __attribute__((ext_vector_type(2))) float v2f;
typedef __attribute__((ext_vector_type(4))) float v4f;
typedef __attribute__((ext_vector_type(8))) float v8f;
typedef __attribute__((ext_vector_type(4))) int   v4i;

#define DIM 1024
#define SEQ 2048
#define BATCH 2
#define NHEADS 16
#define HDIM 64
#define MTOT (BATCH * SEQ)   // 4096 rows

// ---- CDNA5 async global->LDS staging (ASYNCcnt path), with fallbacks ------
#if __has_builtin(__builtin_amdgcn_global_load_async_to_lds_b128)
#define ASYNC_B128 1
#else
#define ASYNC_B128 0
#endif
#if __has_builtin(__builtin_amdgcn_global_load_async_to_lds_b32)
#define ASYNC_B32 1
#else
#define ASYNC_B32 0
#endif

__device__ __forceinline__ void cp16_async(void* lds, const void* g) {
#if ASYNC_B128
  __builtin_amdgcn_global_load_async_to_lds_b128(
      (__attribute__((address_space(1))) v4i*)g,
      (__attribute__((address_space(3))) v4i*)lds, 0, 0);
#else
  *(v4f*)lds = *(const v4f*)g;
#endif
}

__device__ __forceinline__ void cp4_async(void* lds, const void* g) {
#if ASYNC_B32
  __builtin_amdgcn_global_load_async_to_lds_b32(
      (__attribute__((address_space(1))) int*)g,
      (__attribute__((address_space(3))) int*)lds, 0, 0);
#else
  *(float*)lds = *(const float*)g;
#endif
}

__device__ __forceinline__ void async_wait_all() {
#if ASYNC_B128 || ASYNC_B32
#if __has_builtin(__builtin_amdgcn_s_wait_asynccnt)
  __builtin_amdgcn_s_wait_asynccnt(0);
#else
  asm volatile("s_wait_asynccnt 0x0" ::: "memory");
#endif
#endif
}

// ---------------------------------------------------------------------------
// GEMM + bias: out[M,N] = A[M,K] @ W[K,N] + bias[N]
// 256 threads = 8 waves; 64x64 tile per workgroup.
// Wave w: rows [(w>>1)*16,+16), cols [(w&1)*32,+32) -> two 16x16 f32 accs.
// A staged row(k)-major (async B128); W staged TRANSPOSED [n][k] via async
// B32 gathers so every WMMA fragment is one contiguous ds_load_b64.
// ---------------------------------------------------------------------------
__global__ __launch_bounds__(256) void gemm_bias_f32(
    const float* __restrict__ A, const float* __restrict__ W,
    const float* __restrict__ bias, float* __restrict__ out,
    int M, int N, int K) {
  __shared__ alignas(16) float Xs[64][36];  // [row][k]   stride 144B (16B mult)
  __shared__ alignas(16) float Wt[64][36];  // [n][k]     transposed

  const int tid  = threadIdx.x;
  const int wave = tid >> 5;
  const int lane = tid & 31;
  const int lm   = lane & 15;
  const int half = lane >> 4;
  const int wr   = wave >> 1;        // 0..3 : 16-row strip
  const int wc   = wave & 1;         // 0..1 : 32-col strip

  const int tileN = blockIdx.x * 64;
  const int tileM = blockIdx.y * 64;

  v8f acc0 = {};
  v8f acc1 = {};

  for (int k0 = 0; k0 < K; k0 += 32) {
    __syncthreads();
    // A tile: 64 rows x 32 k, 16-byte chunks (512 chunks / 256 thr)
    for (int i = tid; i < 64 * 8; i += 256) {
      int r = i >> 3, c = (i & 7) << 2;
      cp16_async(&Xs[r][c], &A[(size_t)(tileM + r) * K + (k0 + c)]);
    }
    // W tile transposed: per-element async gather (2048 elems / 256 thr)
    for (int i = tid; i < 32 * 64; i += 256) {
      int r = i >> 6, c = i & 63;           // r = k, c = n
      cp4_async(&Wt[c][r], &W[(size_t)(k0 + r) * N + (tileN + c)]);
    }
    async_wait_all();
    __syncthreads();

    const int ar  = wr * 16 + lm;
    const int bn0 = wc * 32 + lm;
#pragma unroll
    for (int kk = 0; kk < 32; kk += 4) {
      v2f a  = *(const v2f*)&Xs[ar][kk + 2 * half];
      v2f b0 = *(const v2f*)&Wt[bn0][kk + 2 * half];
      v2f b1 = *(const v2f*)&Wt[bn0 + 16][kk + 2 * half];
      acc0 = __builtin_amdgcn_wmma_f32_16x16x4_f32(false, a, false, b0,
                                                   (short)0, acc0, false, false);
      acc1 = __builtin_amdgcn_wmma_f32_16x16x4_f32(false, a, false, b1,
                                                   (short)0, acc1, false, false);
    }
  }

  const int orow = tileM + wr * 16;
  const int ocol = tileN + wc * 32;
#pragma unroll
  for (int v = 0; v < 8; ++v) {
    int r = orow + v + 8 * half;
    out[(size_t)r * N + ocol + lm]      = acc0[v] + bias[ocol + lm];
    out[(size_t)r * N + ocol + 16 + lm] = acc1[v] + bias[ocol + 16 + lm];
  }
}

// ---------------------------------------------------------------------------
// Flash-style attention per (batch, head).
// Grid: (32 q-blocks of 64 rows, 32 batch*head). Block: 128 thr = 4 waves.
// Each wave owns 16 query rows. K tile staged row-major (async B128);
// V tile staged TRANSPOSED [dim][key] (async B32) so the P@V B-fragments are
// contiguous b64 loads. Online softmax via 16-lane shfl_xor reductions.
// ---------------------------------------------------------------------------
__global__ __launch_bounds__(128) void attn_f32(
    const float* __restrict__ Q, const float* __restrict__ K,
    const float* __restrict__ V, float* __restrict__ CTX) {
  __shared__ alignas(16) float Ks[16][68];  // [key][dim]  stride 272B (16B mult)
  __shared__ alignas(16) float Vt[64][20];  // [dim][key]  transposed
  __shared__ alignas(16) float Ps[4][16][20];

  const int tid  = threadIdx.x;
  const int wave = tid >> 5;
  const int lane = tid & 31;
  const int lm   = lane & 15;
  const int half = lane >> 4;

  const int qblk = blockIdx.x;          // 0..31
  const int bh   = blockIdx.y;          // 0..31
  const int b    = bh >> 4;
  const int h    = bh & 15;

  const int q0 = qblk * 64 + wave * 16;

  // Preload this wave's Q rows as 16 A-fragments (whole head_dim=64).
  const size_t baseQ = ((size_t)b * SEQ + q0 + lm) * DIM + h * HDIM;
  v2f aq[16];
#pragma unroll
  for (int s = 0; s < 16; ++s)
    aq[s] = *(const v2f*)&Q[baseQ + 4 * s + 2 * half];

  v8f o0 = {}, o1 = {}, o2 = {}, o3 = {};
  float mrow[8], lrow[8];
#pragma unroll
  for (int v = 0; v < 8; ++v) { mrow[v] = -1e30f; lrow[v] = 0.0f; }

  const size_t baseKV = (size_t)b * SEQ * DIM + h * HDIM;

  for (int j = 0; j < SEQ / 16; ++j) {
    __syncthreads();
    // K tile: 16 keys x 64 dims, 16B chunks (256 chunks / 128 thr)
    for (int i = tid; i < 16 * 16; i += 128) {
      int r = i >> 4, c = (i & 15) << 2;
      cp16_async(&Ks[r][c], &K[baseKV + (size_t)(j * 16 + r) * DIM + c]);
    }
    // V tile transposed: per-element async gather (1024 elems / 128 thr)
    for (int i = tid; i < 16 * 64; i += 128) {
      int r = i >> 6, c = i & 63;          // r = key, c = dim
      cp4_async(&Vt[c][r], &V[baseKV + (size_t)(j * 16 + r) * DIM + c]);
    }
    async_wait_all();
    __syncthreads();

    // S(16x16) = Q(16x64) @ K_tile^T(64x16)
    v8f s = {};
#pragma unroll
    for (int sk = 0; sk < 16; ++sk) {
      v2f bf = *(const v2f*)&Ks[lm][4 * sk + 2 * half];
      s = __builtin_amdgcn_wmma_f32_16x16x4_f32(false, aq[sk], false, bf,
                                                (short)0, s, false, false);
    }

    // online softmax update per row (row = v + 8*half, cols across 16 lanes)
#pragma unroll
    for (int v = 0; v < 8; ++v) {
      float val = s[v] * 0.125f;   // 1/sqrt(64)
      float rmax = val;
      rmax = fmaxf(rmax, __shfl_xor(rmax, 1, 16));
      rmax = fmaxf(rmax, __shfl_xor(rmax, 2, 16));
      rmax = fmaxf(rmax, __shfl_xor(rmax, 4, 16));
      rmax = fmaxf(rmax, __shfl_xor(rmax, 8, 16));
      float mnew = fmaxf(mrow[v], rmax);
      float p = __expf(val - mnew);
      float rsum = p;
      rsum += __shfl_xor(rsum, 1, 16);
      rsum += __shfl_xor(rsum, 2, 16);
      rsum += __shfl_xor(rsum, 4, 16);
      rsum += __shfl_xor(rsum, 8, 16);
      float corr = __expf(mrow[v] - mnew);
      lrow[v] = lrow[v] * corr + rsum;
      mrow[v] = mnew;
      o0[v] *= corr; o1[v] *= corr; o2[v] *= corr; o3[v] *= corr;
      Ps[wave][v + 8 * half][lm] = p;   // C-layout -> LDS for A-layout reload
    }

    // ctx(16x64) += P(16x16) @ V_tile(16x64)   (B[kk][n] = Vt[n][kk])
#pragma unroll
    for (int sk = 0; sk < 4; ++sk) {
      const int kr = 4 * sk + 2 * half;
      v2f a  = *(const v2f*)&Ps[wave][lm][kr];
      v2f b0 = *(const v2f*)&Vt[lm][kr];
      v2f b1 = *(const v2f*)&Vt[16 + lm][kr];
      v2f b2 = *(const v2f*)&Vt[32 + lm][kr];
      v2f b3 = *(const v2f*)&Vt[48 + lm][kr];
      o0 = __builtin_amdgcn_wmma_f32_16x16x4_f32(false, a, false, b0, (short)0, o0, false, false);
      o1 = __builtin_amdgcn_wmma_f32_16x16x4_f32(false, a, false, b1, (short)0, o1, false, false);
      o2 = __builtin_amdgcn_wmma_f32_16x16x4_f32(false, a, false, b2, (short)0, o2, false, false);
      o3 = __builtin_amdgcn_wmma_f32_16x16x4_f32(false, a, false, b3, (short)0, o3, false, false);
    }
  }

  // normalize and write ctx (flat layout: (b*T + t)*DIM + h*64 + d)
  const size_t baseO = ((size_t)b * SEQ + q0) * DIM + h * HDIM;
#pragma unroll
  for (int v = 0; v < 8; ++v) {
    float inv = 1.0f / lrow[v];
    size_t r = baseO + (size_t)(v + 8 * half) * DIM;
    CTX[r + lm]      = o0[v] * inv;
    CTX[r + 16 + lm] = o1[v] * inv;
    CTX[r + 32 + lm] = o2[v] * inv;
    CTX[r + 48 + lm] = o3[v] * inv;
  }
}

// ---------------------------------------------------------------------------
extern "C" void kernel_launch(void* const* d_in, const int* in_sizes, int n_in,
                              void* d_out, int out_size, void* d_ws, size_t ws_size,
                              hipStream_t stream) {
  const float* x  = (const float*)d_in[0];
  const float* Wq = (const float*)d_in[1];
  const float* bq = (const float*)d_in[2];
  const float* Wk = (const float*)d_in[3];
  const float* bk = (const float*)d_in[4];
  const float* Wv = (const float*)d_in[5];
  const float* bv = (const float*)d_in[6];
  const float* Wo = (const float*)d_in[7];
  const float* bo = (const float*)d_in[8];
  float* out = (float*)d_out;

  const size_t mat = (size_t)MTOT * DIM;  // 4096*1024 floats
  float* Qb = (float*)d_ws;
  float* Kb = Qb + mat;
  float* Vb = Kb + mat;
  float* Cb = Vb + mat;

  dim3 ggrid(DIM / 64, MTOT / 64);   // (16, 64)
  dim3 gblk(256);
  gemm_bias_f32<<<ggrid, gblk, 0, stream>>>(x, Wq, bq, Qb, MTOT, DIM, DIM);
  gemm_bias_f32<<<ggrid, gblk, 0, stream>>>(x, Wk, bk, Kb, MTOT, DIM, DIM);
  gemm_bias_f32<<<ggrid, gblk, 0, stream>>>(x, Wv, bv, Vb, MTOT, DIM, DIM);

  dim3 agrid(SEQ / 64, BATCH * NHEADS);  // (32, 32)
  attn_f32<<<agrid, dim3(128), 0, stream>>>(Qb, Kb, Vb, Cb);

  gemm_bias_f32<<<ggrid, gblk, 0, stream>>>(Cb, Wo, bo, out, MTOT, DIM, DIM);
}